// TestModel_31610959299255
// MI455X (gfx1250) — compile-verified
//
#include <hip/hip_runtime.h>

// ---------------------------------------------------------------------------
// Elman-style affine recurrence h_{t+1} = x_t@ih + h_t@hh + bias, T=65536,
// B=2, DIN=4, DH=5; output = sum(h_T).
//
// Strategy (MI455X): the recurrence is linear => associative scan.
//   Phase 0: precompute W[44][16] = [hh^8 ; ih*hh^7 ; ... ; ih ; bias*Sum(hh^r)]
//            (K-padded, col-padded) and M128 = hh^128, single scalar thread.
//   Phase 1: 512 chunks x 128 steps, one wave32 per chunk. Each 8-step block is
//            h_{t+8} = [h(5)|pad(3)|x_t..x_{t+7}(32)|1|pad(3)] (16x44) @ W (44x16)
//            = 11 accumulate-chained v_wmma_f32_16x16x4_f32. State is fed back
//            into A-layout with 3 branchless ds_bpermute per 8 steps.
//   Phase 2: combine 512 chunk results with h = h@hh^128 + S_c (scalar, tiny),
//            write sum(h) to d_out.
// ---------------------------------------------------------------------------

#define T_STEPS 65536
#define BB      2
#define DIN     4
#define DH      5
#define R_STEPS 8
#define CHUNK   128
#define NBLK    (CHUNK / R_STEPS)   // 16
#define NCHUNK  (T_STEPS / CHUNK)   // 512
#define NSLICE  11                  // K = 44 = 11 * 4

// workspace layout (float offsets)
#define WS_WCOEF 0                  // [44][16]
#define WS_M128  704                // [5][5]
#define WS_S     736                // [512][2][5]

typedef __attribute__((ext_vector_type(2))) float v2f;
typedef __attribute__((ext_vector_type(8))) float v8f;

static __device__ inline v8f wmma4(v2f a, v2f b, v8f c) {
  // (neg_a, A, neg_b, B, c_mod, C, reuse_a, reuse_b) -- probed 8-arg f32 form
  return __builtin_amdgcn_wmma_f32_16x16x4_f32(false, a, false, b, (short)0, c,
                                               false, false);
}

static __device__ inline float bperm_f(int byte_idx, float v) {
  return __int_as_float(
      __builtin_amdgcn_ds_bpermute(byte_idx, __float_as_int(v)));
}

// ------------------------- Phase 0: precompute ------------------------------
__global__ void rnn_precompute_kernel(const float* __restrict__ ih,
                                      const float* __restrict__ hh,
                                      const float* __restrict__ bias,
                                      float* __restrict__ ws) {
  if (threadIdx.x != 0 || blockIdx.x != 0) return;

  float pw[R_STEPS + 1][DH][DH];  // pw[k] = hh^k
  for (int i = 0; i < DH; ++i)
    for (int j = 0; j < DH; ++j) pw[0][i][j] = (i == j) ? 1.0f : 0.0f;
  for (int k = 1; k <= R_STEPS; ++k)
    for (int i = 0; i < DH; ++i)
      for (int j = 0; j < DH; ++j) {
        float acc = 0.0f;
        for (int m = 0; m < DH; ++m) acc += pw[k - 1][i][m] * hh[m * DH + j];
        pw[k][i][j] = acc;
      }

  // geometric sum Sum_{k=0..7} hh^k (for the folded bias row)
  float sg[DH][DH];
  for (int i = 0; i < DH; ++i)
    for (int j = 0; j < DH; ++j) {
      float acc = 0.0f;
      for (int k = 0; k < R_STEPS; ++k) acc += pw[k][i][j];
      sg[i][j] = acc;
    }

  float* wc = ws + WS_WCOEF;
  for (int i = 0; i < 44 * 16; ++i) wc[i] = 0.0f;

  // rows 0..4 : hh^8  (h carry-over)
  for (int i = 0; i < DH; ++i)
    for (int j = 0; j < DH; ++j) wc[i * 16 + j] = pw[R_STEPS][i][j];
  // rows 8+4r+j : (ih @ hh^{7-r})[j]  (x_{t+r} contribution)
  for (int r = 0; r < R_STEPS; ++r)
    for (int j = 0; j < DIN; ++j)
      for (int nn = 0; nn < DH; ++nn) {
        float acc = 0.0f;
        for (int k = 0; k < DH; ++k)
          acc += ih[j * DH + k] * pw[R_STEPS - 1 - r][k][nn];
        wc[(8 + 4 * r + j) * 16 + nn] = acc;
      }
  // row 40 : bias @ Sum(hh^r)
  for (int nn = 0; nn < DH; ++nn) {
    float acc = 0.0f;
    for (int k = 0; k < DH; ++k) acc += bias[k] * sg[k][nn];
    wc[40 * 16 + nn] = acc;
  }

  // M128 = (hh^8)^16 via 4 squarings
  float a[DH][DH], tmp[DH][DH];
  for (int i = 0; i < DH; ++i)
    for (int j = 0; j < DH; ++j) a[i][j] = pw[R_STEPS][i][j];
  for (int sq = 0; sq < 4; ++sq) {
    for (int i = 0; i < DH; ++i)
      for (int j = 0; j < DH; ++j) {
        float acc = 0.0f;
        for (int k = 0; k < DH; ++k) acc += a[i][k] * a[k][j];
        tmp[i][j] = acc;
      }
    for (int i = 0; i < DH; ++i)
      for (int j = 0; j < DH; ++j) a[i][j] = tmp[i][j];
  }
  for (int i = 0; i < DH; ++i)
    for (int j = 0; j < DH; ++j) ws[WS_M128 + i * DH + j] = a[i][j];
}

// ------------------------- Phase 1: chunk scan ------------------------------
__global__ void __launch_bounds__(32)
rnn_chunk_kernel(const float* __restrict__ x,
                 const float* __restrict__ ws,
                 float* __restrict__ S) {
  const int c    = blockIdx.x;           // chunk id, 0..511
  const int lane = threadIdx.x;          // 0..31 (one wave32)
  const int n    = lane & 15;            // row index (A) / col index (B,D)
  const int koff = (lane < 16) ? 0 : 2;  // K offset within a 16x4 A fragment
  const float colmask = (n < BB) ? 1.0f : 0.0f;  // only rows 0..1 are live
  const float merge_sel = (lane & 8) ? 1.0f : 0.0f;  // row1 if lane's bit3 set

  // B fragments: W (44x16) sliced into 11 fragments of 4x16.
  const float* wc = ws + WS_WCOEF;
  v2f bfrag[NSLICE];
#pragma unroll
  for (int s = 0; s < NSLICE; ++s) {
    v2f bf;
    bf.x = wc[(4 * s + koff) * 16 + n];
    bf.y = wc[(4 * s + koff + 1) * 16 + n];
    bfrag[s] = bf;
  }

  // A state fragments: slice0 = h[:,0..3], slice1 = h[:,4]|pad. h0 = 0.
  v2f a0 = {0.0f, 0.0f};
  v2f a1 = {0.0f, 0.0f};
  v2f a_one;  // K=40 "ones" column multiplying the folded bias row
  a_one.x = (lane < 16 && n < BB) ? 1.0f : 0.0f;
  a_one.y = 0.0f;

  // bpermute byte indices for the state feedback: source lane = col + 8*b
  // (b = this lane's A-row = n); indices for n>=2 wrap harmlessly (masked).
  const int ib0 = 4 * (8 * n + koff);      // h_new[n][koff]
  const int ib1 = 4 * (8 * n + koff + 1);  // h_new[n][koff+1]
  const int ib2 = 4 * (8 * n + 4);         // h_new[n][4]

  v8f d = {0.f, 0.f, 0.f, 0.f, 0.f, 0.f, 0.f, 0.f};
  const int t0 = c * CHUNK;

  for (int blk = 0; blk < NBLK; ++blk) {
    const int tb = t0 + blk * R_STEPS;

    // prefetch next block's x stream (global_prefetch), clamped in-bounds
    int tpf = tb + R_STEPS;
    if (tpf > T_STEPS - 1) tpf = T_STEPS - 1;
    __builtin_prefetch(x + (size_t)tpf * (BB * DIN), 0, 1);

    v8f acc = {0.f, 0.f, 0.f, 0.f, 0.f, 0.f, 0.f, 0.f};
    acc = wmma4(a0, bfrag[0], acc);     // h[:,0..3] * hh^8 rows 0..3
    acc = wmma4(a1, bfrag[1], acc);     // h[:,4]    * hh^8 row 4
#pragma unroll
    for (int r = 0; r < R_STEPS; ++r) {
      // A fragment for x_{tb+r}: lane(b=n) supplies x[t][b][koff..koff+1]
      int idx = (n < BB) ? ((tb + r) * (BB * DIN) + n * DIN + koff) : 0;
      v2f xr = *(const v2f*)(x + idx);
      xr.x *= colmask;
      xr.y *= colmask;
      acc = wmma4(xr, bfrag[2 + r], acc);
    }
    acc = wmma4(a_one, bfrag[10], acc); // + bias * Sum(hh^r)
    d = acc;

    // ---- state feedback: D rows 0..1 -> A-layout, fully branchless ----
    // D rows 2..15 are exactly 0 (A rows >= 2 are kept 0), so merge rows
    // 0 and 1 into one register: lane s in [0,8) holds row0 col s, lane s
    // in [8,16) holds row1 col s (zero pad). Pure v_cndmask, EXEC stays all-1s.
    const float d0 = d[0], d1 = d[1];
    const float merged = merge_sel * d1 + (1.0f - merge_sel) * d0;
    float g0 = bperm_f(ib0, merged);
    float g1 = bperm_f(ib1, merged);
    float g2 = bperm_f(ib2, merged);
    a0.x = g0 * colmask;
    a0.y = g1 * colmask;
    a1.x = (lane < 16) ? (g2 * colmask) : 0.0f;
    a1.y = 0.0f;
  }

  // Store S_c (2x5) from the final accumulator.
  if (lane < DH) {
    S[c * (BB * DH) + lane]      = d[0];
    S[c * (BB * DH) + DH + lane] = d[1];
  }
}

// ------------------------- Phase 2: combine ---------------------------------
__global__ void rnn_combine_kernel(const float* __restrict__ ws,
                                   float* __restrict__ out) {
  if (threadIdx.x != 0 || blockIdx.x != 0) return;
  const float* M = ws + WS_M128;
  const float* S = ws + WS_S;
  float h[BB][DH] = {};
  for (int c = 0; c < NCHUNK; ++c) {
    float nh[BB][DH];
    for (int b = 0; b < BB; ++b)
      for (int j = 0; j < DH; ++j) {
        float acc = S[c * (BB * DH) + b * DH + j];
        for (int k = 0; k < DH; ++k) acc += h[b][k] * M[k * DH + j];
        nh[b][j] = acc;
      }
    for (int b = 0; b < BB; ++b)
      for (int j = 0; j < DH; ++j) h[b][j] = nh[b][j];
  }
  float s = 0.0f;
  for (int b = 0; b < BB; ++b)
    for (int j = 0; j < DH; ++j) s += h[b][j];
  out[0] = s;
}

// ---------------------------------------------------------------------------
extern "C" void kernel_launch(void* const* d_in, const int* in_sizes, int n_in,
                              void* d_out, int out_size, void* d_ws,
                              size_t ws_size, hipStream_t stream) {
  (void)in_sizes; (void)n_in; (void)out_size; (void)ws_size;
  const float* x    = (const float*)d_in[0];
  const float* ih   = (const float*)d_in[1];
  const float* hh   = (const float*)d_in[2];
  const float* bias = (const float*)d_in[3];
  float* ws = (float*)d_ws;

  rnn_precompute_kernel<<<1, 32, 0, stream>>>(ih, hh, bias, ws);
  rnn_chunk_kernel<<<NCHUNK, 32, 0, stream>>>(x, ws, ws + WS_S);
  rnn_combine_kernel<<<1, 32, 0, stream>>>(ws, (float*)d_out);
}